// Encoder_42056319762462
// MI455X (gfx1250) — compile-verified
//
#include <hip/hip_runtime.h>
#include <hip/hip_bf16.h>
#include <stdint.h>

#define DFEAT 256
#define WPR_MAX 320   // words per bitmap row, max (N <= 10240)

typedef __attribute__((ext_vector_type(2))) float v2f;
typedef __attribute__((ext_vector_type(8))) float v8f;

// ---------------- utility: zero a u32 region ----------------
__global__ void zero_u32(uint32_t* __restrict__ p, long n) {
  long i = (long)blockIdx.x * blockDim.x + threadIdx.x;
  long stride = (long)gridDim.x * blockDim.x;
  for (; i < n; i += stride) p[i] = 0u;
}

// ---------------- build deduped adjacency bitmap + degrees ----------------
__global__ void build_adj(const int* __restrict__ ei, int E, int n, int wpr,
                          uint32_t* __restrict__ bitmap,
                          uint32_t* __restrict__ deg_r,
                          uint32_t* __restrict__ deg_c) {
  int e = blockIdx.x * blockDim.x + threadIdx.x;
  if (e >= E) return;
  int r = ei[e];
  int c = ei[E + e];
  if ((unsigned)r >= (unsigned)n || (unsigned)c >= (unsigned)n) return;
  uint32_t bit = 1u << (c & 31);
  uint32_t old = atomicOr(&bitmap[(size_t)r * wpr + (c >> 5)], bit);
  if (!(old & bit)) {          // first time this (r,c) appears -> count once
    atomicAdd(&deg_r[r], 1u);
    atomicAdd(&deg_c[c], 1u);
  }
}

__global__ void make_dinv(const uint32_t* __restrict__ deg_r,
                          const uint32_t* __restrict__ deg_c,
                          float* __restrict__ dinv_r,
                          float* __restrict__ dinv_c, int n) {
  int i = blockIdx.x * blockDim.x + threadIdx.x;
  if (i >= n) return;
  uint32_t dr = deg_r[i], dc = deg_c[i];
  dinv_r[i] = dr ? rsqrtf((float)dr) : 0.0f;
  dinv_c[i] = dc ? rsqrtf((float)dc) : 0.0f;
}

// ---------------- C[M x 256] = A[M x 256] * B[256 x 256], fp32 WMMA ----------------
// block = 256 threads = 8 waves; each wave owns one 16x16 C tile.
// grid = (M/16, 256/128)
__global__ void __launch_bounds__(256)
gemm256_wmma(const float* __restrict__ A, const float* __restrict__ B,
             float* __restrict__ C) {
  const int lane = threadIdx.x & 31;
  const int wave = threadIdx.x >> 5;
  const int l16  = lane & 15;
  const int lh   = lane >> 4;                  // 0: K pair {0,1}, 1: K pair {2,3}
  const int m0   = blockIdx.x * 16;
  const int n0   = blockIdx.y * 128 + wave * 16;

  const float* arow = A + (size_t)(m0 + l16) * DFEAT;   // A-frag row for this lane
  const float* bcol = B + n0 + l16;                     // B-frag column for this lane

  v8f acc = {};
#pragma unroll 4
  for (int k = 0; k < DFEAT; k += 4) {
    const int kk = k + 2 * lh;
    // A 16x4 fp32 frag: VGPR0=K@(kk), VGPR1=K@(kk+1)  (8B aligned load)
    v2f a = *(const v2f*)(arow + kk);
    // B 4x16 fp32 frag: rows striped across lanes, same K-pair split as A
    v2f b;
    b.x = bcol[(size_t)kk * DFEAT];
    b.y = bcol[(size_t)(kk + 1) * DFEAT];
    acc = __builtin_amdgcn_wmma_f32_16x16x4_f32(
        /*neg_a=*/false, a, /*neg_b=*/false, b,
        /*c_mod=*/(short)0, acc, /*reuse_a=*/false, /*reuse_b=*/false);
  }

  // C/D 16x16 fp32 layout: VGPR v -> row m0 + v + 8*lh, col n0 + l16
  float* crow = C + (size_t)(m0 + 8 * lh) * DFEAT + n0 + l16;
#pragma unroll
  for (int v = 0; v < 8; ++v) crow[(size_t)v * DFEAT] = acc[v];
}

// ---------------- fused sparse propagation + bias (+ReLU) ----------------
// lsym==1:  out = Y[r] - dinv[r] * sum_{j in row r} dinv[j]*Y[j]  + bias   (L = I - Dr^-1/2 A Dr^-1/2)
// lsym==0:  out =        dinv[r] * sum_{j in row r} dinv[j]*Y[j]  + bias   (Dc^-1/2 A Dc^-1/2)
// one block per row; thread f owns feature f; bit scan is block-uniform (no divergence).
__global__ void __launch_bounds__(256)
spmm_prop(const uint32_t* __restrict__ bitmap, int wpr,
          const float* __restrict__ Y,
          const float* __restrict__ dinv,
          const float* __restrict__ bias,
          float* __restrict__ out0,
          float* __restrict__ out1,
          int lsym, int do_relu) {
  __shared__ uint32_t wbits[WPR_MAX];
  const int r = blockIdx.x;
  const int f = threadIdx.x;

  for (int i = threadIdx.x; i < wpr; i += blockDim.x)
    wbits[i] = bitmap[(size_t)r * wpr + i];
  __syncthreads();

  float acc = 0.0f;
  for (int i = 0; i < wpr; ++i) {
    uint32_t bits = wbits[i];
    const int base = i << 5;
    while (bits) {
      const int j = base + __ffs(bits) - 1;
      bits &= bits - 1;
      acc = fmaf(dinv[j], Y[(size_t)j * DFEAT + f], acc);  // Y row read: fully coalesced
    }
  }

  float v;
  if (lsym) v = Y[(size_t)r * DFEAT + f] - dinv[r] * acc + bias[f];
  else      v = dinv[r] * acc + bias[f];
  if (do_relu) v = fmaxf(v, 0.0f);

  out0[(size_t)r * DFEAT + f] = v;
  if (out1) out1[(size_t)r * DFEAT + f] = v;
}

// ---------------- host-side orchestration ----------------
extern "C" void kernel_launch(void* const* d_in, const int* in_sizes, int n_in,
                              void* d_out, int out_size, void* d_ws, size_t ws_size,
                              hipStream_t stream) {
  const float* x  = (const float*)d_in[0];
  const int*   ei = (const int*)d_in[1];
  const float* W1 = (const float*)d_in[2];
  const float* b1 = (const float*)d_in[3];
  const float* W2 = (const float*)d_in[4];
  const float* b2 = (const float*)d_in[5];

  const int N   = in_sizes[0] / DFEAT;   // 10000
  const int E   = in_sizes[1] / 2;       // 320000
  const int wpr = (N + 31) >> 5;         // 313

  float* out = (float*)d_out;

  // workspace carve-out (256B aligned chunks)
  char* ws = (char*)d_ws;
  size_t off = 0;
  auto alloc = [&](size_t bytes) -> char* {
    char* p = ws + off;
    off += (bytes + 255) & ~(size_t)255;
    return p;
  };
  uint32_t* bitmap = (uint32_t*)alloc((size_t)N * wpr * sizeof(uint32_t)); // 12.5 MB
  uint32_t* deg_r  = (uint32_t*)alloc((size_t)N * sizeof(uint32_t));
  uint32_t* deg_c  = (uint32_t*)alloc((size_t)N * sizeof(uint32_t));
  float*    dinv_r = (float*)alloc((size_t)N * sizeof(float));
  float*    dinv_c = (float*)alloc((size_t)N * sizeof(float));
  float*    XW1    = (float*)alloc((size_t)N * DFEAT * sizeof(float));
  float*    h1     = (float*)alloc((size_t)N * DFEAT * sizeof(float));
  float*    h2     = (float*)alloc((size_t)N * DFEAT * sizeof(float));
  float*    G1     = (float*)alloc((size_t)N * DFEAT * sizeof(float));
  float*    G2     = (float*)alloc((size_t)N * DFEAT * sizeof(float));
  (void)ws_size; (void)n_in; (void)out_size;

  // 1) reset adjacency state (must happen every call: deterministic re-run)
  zero_u32<<<1024, 256, 0, stream>>>(bitmap, (long)N * wpr);
  zero_u32<<<(N + 255) / 256, 256, 0, stream>>>(deg_r, N);
  zero_u32<<<(N + 255) / 256, 256, 0, stream>>>(deg_c, N);

  // 2) deduped adjacency bitmap + exact row/col degrees
  build_adj<<<(E + 255) / 256, 256, 0, stream>>>(ei, E, N, wpr, bitmap, deg_r, deg_c);

  // 3) D^-1/2 factors
  make_dinv<<<(N + 255) / 256, 256, 0, stream>>>(deg_r, deg_c, dinv_r, dinv_c, N);

  // 4) XW1 = x @ W1   (fp32 WMMA)
  dim3 ggrid(N / 16, DFEAT / 128);
  gemm256_wmma<<<ggrid, 256, 0, stream>>>(x, W1, XW1);

  // 5) layer-1 propagation + bias + ReLU
  spmm_prop<<<N, 256, 0, stream>>>(bitmap, wpr, XW1, dinv_r, b1, h1, nullptr, /*lsym=*/1, /*relu=*/1);
  spmm_prop<<<N, 256, 0, stream>>>(bitmap, wpr, XW1, dinv_c, b1, h2, nullptr, /*lsym=*/0, /*relu=*/1);

  // 6) layer-2 dense parts
  gemm256_wmma<<<ggrid, 256, 0, stream>>>(h1, W2, G1);
  gemm256_wmma<<<ggrid, 256, 0, stream>>>(h2, W2, G2);

  // 7) layer-2 propagation + bias -> outputs (z, z1, z2) with z == z2
  float* z  = out;
  float* z1 = out + (size_t)N * DFEAT;
  float* z2 = out + 2 * (size_t)N * DFEAT;
  spmm_prop<<<N, 256, 0, stream>>>(bitmap, wpr, G1, dinv_r, b2, z1, nullptr, /*lsym=*/1, /*relu=*/0);
  spmm_prop<<<N, 256, 0, stream>>>(bitmap, wpr, G2, dinv_c, b2, z2, z,      /*lsym=*/0, /*relu=*/0);
}